// Adaptive_FP_75161927680023
// MI455X (gfx1250) — compile-verified
//
#include <hip/hip_runtime.h>
#include <hip/hip_bf16.h>

// Problem constants (compile-time, from the reference)
#define BB     8
#define NPTS   2048
#define MPTS   8192
#define CPTS   64
#define DMODEL 64
#define KNN    16

typedef __attribute__((ext_vector_type(2))) float v2f;
typedef __attribute__((ext_vector_type(8))) float v8f;

// ---------------------------------------------------------------------------
// Kernel 1: per-query top-16 nearest neighbors + gather of their coordinates.
// One thread per query point m; xyz[b] staged in LDS as SoA (24 KB).
// The scan index n is wave-uniform -> LDS reads are broadcasts (bank-conflict
// free). Guarded insertion sort keeps the expected per-candidate cost ~O(1).
// Strict '<' comparisons reproduce stable-argsort tie ordering because
// candidates arrive in increasing index order.
// ---------------------------------------------------------------------------
__global__ __launch_bounds__(256) void knn_topk_kernel(
    const float* __restrict__ xyz,      // [B, N, 3]
    const float* __restrict__ xyz_fp,   // [B, M, 3]
    float* __restrict__ knn_out)        // [B, M, K, 3]
{
    __shared__ float sx[NPTS];
    __shared__ float sy[NPTS];
    __shared__ float sz[NPTS];

    const int b    = blockIdx.x >> 5;          // 32 blocks per batch (8192/256)
    const int mblk = blockIdx.x & 31;
    const int m    = mblk * 256 + threadIdx.x;

    // Stage xyz[b] into LDS (SoA)
    const float* xb = xyz + (size_t)b * NPTS * 3;
    for (int g = threadIdx.x; g < NPTS * 3; g += 256) {
        float v = xb[g];
        int n = g / 3;
        int c = g - 3 * n;
        if (c == 0)      sx[n] = v;
        else if (c == 1) sy[n] = v;
        else             sz[n] = v;
    }
    __syncthreads();

    const float* q = xyz_fp + ((size_t)b * MPTS + m) * 3;
    const float qx = q[0], qy = q[1], qz = q[2];

    float dist[KNN];
    int   idx[KNN];
#pragma unroll
    for (int j = 0; j < KNN; ++j) { dist[j] = 3.0e38f; idx[j] = 0; }

    for (int n = 0; n < NPTS; ++n) {
        float dx = qx - sx[n];
        float dy = qy - sy[n];
        float dz = qz - sz[n];
        float d2 = fmaf(dx, dx, fmaf(dy, dy, dz * dz));
        if (d2 < dist[KNN - 1]) {            // guard: rare after warmup
            float dc = d2;
            int   ic = n;
#pragma unroll
            for (int j = 0; j < KNN; ++j) {
                if (dc < dist[j]) {
                    float tf = dist[j]; dist[j] = dc; dc = tf;
                    int   ti = idx[j];  idx[j]  = ic; ic = ti;
                }
            }
        }
    }

    // Gather neighbor coordinates straight from LDS
    float* o = knn_out + ((size_t)b * MPTS + m) * (KNN * 3);
#pragma unroll
    for (int j = 0; j < KNN; ++j) {
        int n = idx[j];
        o[3 * j + 0] = sx[n];
        o[3 * j + 1] = sy[n];
        o[3 * j + 2] = sz[n];
    }
}

// ---------------------------------------------------------------------------
// Kernel 2: feats = transpose(features) and x = feats @ W1^T + b1 via
// V_WMMA_F32_16X16X4_F32 (fp32 path: exact precision match, and the op is
// bandwidth-bound so lower precision buys nothing).
//
// One wave (32 lanes) computes a 16(N-rows) x 64(d_model) output tile:
//   - A fragments (16x4 f32): lane L<16 holds (M=L, K=0) in v0, (M=L, K=1) in
//     v1; lanes 16-31 hold K=2,3 — loaded directly from features[b,c,n]
//     (coalesced over n), and also stored back as the transpose output.
//   - B fragments (4x16 f32): mirrored layout, read from W1 staged in LDS.
//   - C/D (16x16 f32): VGPR r <-> row M=r (lanes 0-15) / M=r+8 (lanes 16-31),
//     lane <-> column N; accumulator pre-seeded with the bias.
// ---------------------------------------------------------------------------
__global__ __launch_bounds__(256) void transpose_linear_wmma_kernel(
    const float* __restrict__ features, // [B, C=64, N]
    const float* __restrict__ W1,       // [DMODEL=64, C=64] row-major
    const float* __restrict__ b1,       // [DMODEL]
    float* __restrict__ feats_out,      // [B, N, C]
    float* __restrict__ x_out)          // [B, N, DMODEL]
{
    __shared__ float sW[DMODEL * CPTS];
    __shared__ float sb[DMODEL];

    const int b    = blockIdx.x >> 4;       // 16 blocks per batch (2048/128)
    const int nblk = blockIdx.x & 15;

    for (int g = threadIdx.x; g < DMODEL * CPTS; g += 256) sW[g] = W1[g];
    if (threadIdx.x < DMODEL) sb[threadIdx.x] = b1[threadIdx.x];
    __syncthreads();

    const int wave = threadIdx.x >> 5;      // 8 waves -> 128 N-rows per block
    const int lane = threadIdx.x & 31;
    const int half = lane >> 4;             // 0: lanes 0-15, 1: lanes 16-31
    const int l16  = lane & 15;
    const int n0   = nblk * 128 + wave * 16;

    const float* fb = features + (size_t)b * CPTS * NPTS;

    // Preload all 16 A fragments (K = 4*kk .. 4*kk+3)
    v2f afrag[16];
#pragma unroll
    for (int kk = 0; kk < 16; ++kk) {
        const int c0 = 4 * kk + 2 * half;
        afrag[kk].x = fb[(size_t)(c0 + 0) * NPTS + n0 + l16];
        afrag[kk].y = fb[(size_t)(c0 + 1) * NPTS + n0 + l16];
    }

    // Emit the transpose output (each lane writes a contiguous float pair)
    {
        float* fo = feats_out + ((size_t)b * NPTS + n0 + l16) * CPTS;
#pragma unroll
        for (int kk = 0; kk < 16; ++kk) {
            const int c0 = 4 * kk + 2 * half;
            fo[c0 + 0] = afrag[kk].x;
            fo[c0 + 1] = afrag[kk].y;
        }
    }

    // GEMM: 4 column tiles of 16 d_model outputs, 16 K-steps each
    float* xo = x_out + (size_t)b * NPTS * DMODEL;
#pragma unroll
    for (int j = 0; j < 4; ++j) {
        const int d0   = j * 16;
        const float bias = sb[d0 + l16];
        v8f acc;
#pragma unroll
        for (int r = 0; r < 8; ++r) acc[r] = bias;

#pragma unroll
        for (int kk = 0; kk < 16; ++kk) {
            const int c0 = 4 * kk + 2 * half;
            v2f bfrag;
            // B[K=c][N=d0+l16] = W1^T[c][d] = W1[d][c]
            bfrag.x = sW[(d0 + l16) * CPTS + c0 + 0];
            bfrag.y = sW[(d0 + l16) * CPTS + c0 + 1];
            acc = __builtin_amdgcn_wmma_f32_16x16x4_f32(
                /*neg_a=*/false, afrag[kk],
                /*neg_b=*/false, bfrag,
                /*c_mod=*/(short)0, acc,
                /*reuse_a=*/false, /*reuse_b=*/false);
        }

        // Store D tile
#pragma unroll
        for (int r = 0; r < 8; ++r) {
            const int mrow = r + 8 * half;
            xo[(size_t)(n0 + mrow) * DMODEL + d0 + l16] = acc[r];
        }
    }
}

// ---------------------------------------------------------------------------
// Launcher. d_out layout = concat(feats [B,N,64], knn_xyz [B,M,16,3],
// x [B,N,64]) in the reference's return order.
// ---------------------------------------------------------------------------
extern "C" void kernel_launch(void* const* d_in, const int* in_sizes, int n_in,
                              void* d_out, int out_size, void* d_ws, size_t ws_size,
                              hipStream_t stream) {
    (void)in_sizes; (void)n_in; (void)out_size; (void)d_ws; (void)ws_size;

    const float* xyz      = (const float*)d_in[0];  // [B,N,3]
    const float* xyz_fp   = (const float*)d_in[1];  // [B,M,3]
    const float* features = (const float*)d_in[2];  // [B,C,N]
    // d_in[3] = features_fp (computed-but-unused in reference output)
    const float* W1       = (const float*)d_in[4];  // [D,C]
    const float* b1       = (const float*)d_in[5];  // [D]
    // d_in[6] = k (==16, compile-time constant here)

    float* out       = (float*)d_out;
    float* feats_out = out;                                         // B*N*64
    float* knn_out   = out + (size_t)BB * NPTS * CPTS;              // B*M*16*3
    float* x_out     = knn_out + (size_t)BB * MPTS * KNN * 3;       // B*N*64

    // KNN: one thread per query point; 32 blocks per batch
    knn_topk_kernel<<<BB * (MPTS / 256), 256, 0, stream>>>(xyz, xyz_fp, knn_out);

    // Transpose + linear: 8 waves/block, 16 N-rows per wave -> 128 rows/block
    transpose_linear_wmma_kernel<<<BB * (NPTS / 128), 256, 0, stream>>>(
        features, W1, b1, feats_out, x_out);
}